// RelPartialLearnableMultiHeadAttn_7430293422994
// MI455X (gfx1250) — compile-verified
//
#include <hip/hip_runtime.h>

#define QLEN   1024
#define MLEN   1024
#define KLEN   2048
#define NHEAD  8
#define DHEAD  64
#define DMODEL 512
#define BSZ    4
#define SCALE  0.125f
#define NEGBIG (-3.0e38f)

typedef float v2f __attribute__((ext_vector_type(2)));
typedef float v8f __attribute__((ext_vector_type(8)));

static __device__ __forceinline__ v8f wmma_f32(v2f a, v2f b, v8f c) {
    // D = A(16x4 f32) * B(4x16 f32) + C(16x16 f32), full fp32 precision
    return __builtin_amdgcn_wmma_f32_16x16x4_f32(false, a, false, b, (short)0, c, false, false);
}

static __device__ __forceinline__ float lane_gather(float v, int srcLane) {
    return __int_as_float(__builtin_amdgcn_ds_bpermute(srcLane << 2, __float_as_int(v)));
}

// Async global->LDS DMA (CDNA5): 16B per active lane, tracked by ASYNCcnt.
// Low 32 bits of a generic pointer to LDS are the LDS byte offset.
static __device__ __forceinline__ void async_load_b128(float* lds_ptr, const float* gptr) {
    uint32_t lds_off = (uint32_t)(uintptr_t)lds_ptr;
    asm volatile("global_load_async_to_lds_b128 %0, %1, off"
                 :: "v"(lds_off), "v"(gptr) : "memory");
}
static __device__ __forceinline__ void wait_asynccnt0() {
    asm volatile("s_wait_asynccnt 0" ::: "memory");
}

// ---------------------------------------------------------------------------
// GEMM: C[M,512] @ B[512,N].  A rows may be split across two buffers
// (cat(mems,w)): logical row < catSplit -> A0, else A1[row-catSplit].
// Block: 256 threads (8 waves). Block stages B panel [512 x 64] in LDS via
// async-to-LDS b128 DMA; wave wv computes rows [bx*128+wv*16, +16) x 64 cols.
// Per K-step: 1 global b64 A-frag, 4 LDS B-frag pairs, 4 wmma.
// grid = (M/128, N/64), dynamic LDS = 128 KB.
// ---------------------------------------------------------------------------
__global__ __launch_bounds__(256)
void gemm512(const float* __restrict__ A0, const float* __restrict__ A1,
             int catSplit, const float* __restrict__ B,
             float* __restrict__ C, int N)
{
    extern __shared__ float Bs[];              // [512][64]
    const int tid = threadIdx.x;
    const int n0  = blockIdx.y * 64;

    for (int idx = tid; idx < 512 * 16; idx += 256) {   // 8192 x b128
        int k  = idx >> 4;                     // B row
        int c4 = (idx & 15) << 2;              // col offset (floats)
        async_load_b128(Bs + (size_t)idx * 4, B + (size_t)k * N + n0 + c4);
    }
    wait_asynccnt0();
    __syncthreads();

    const int wave = tid >> 5;
    const int lane = tid & 31;
    const int ln16 = lane & 15;
    const int hi   = lane >> 4;
    const int kb   = hi << 1;                  // K sub-pair 0 or 2
    const int m0   = blockIdx.x * 128 + wave * 16;
    const int row  = m0 + ln16;
    const float* Arow = (row < catSplit) ? (A0 + (size_t)row * DMODEL)
                                         : (A1 + (size_t)(row - catSplit) * DMODEL);

    v8f acc0 = {0.f,0.f,0.f,0.f,0.f,0.f,0.f,0.f};
    v8f acc1 = acc0, acc2 = acc0, acc3 = acc0;

    for (int k0 = 0; k0 < DMODEL; k0 += 4) {
        float2 a2 = *(const float2*)(Arow + k0 + kb);      // contiguous -> b64
        v2f a; a.x = a2.x; a.y = a2.y;
        const float* b0 = Bs + (size_t)(k0 + kb) * 64 + ln16;
        const float* b1 = b0 + 64;                          // next K row
        v2f b;
        b.x = b0[0];  b.y = b1[0];  acc0 = wmma_f32(a, b, acc0);
        b.x = b0[16]; b.y = b1[16]; acc1 = wmma_f32(a, b, acc1);
        b.x = b0[32]; b.y = b1[32]; acc2 = wmma_f32(a, b, acc2);
        b.x = b0[48]; b.y = b1[48]; acc3 = wmma_f32(a, b, acc3);
    }

    #pragma unroll
    for (int v = 0; v < 8; ++v) {
        float* Cp = C + (size_t)(m0 + v + (hi << 3)) * N + n0 + ln16;
        Cp[0]  = acc0[v];
        Cp[16] = acc1[v];
        Cp[32] = acc2[v];
        Cp[48] = acc3[v];
    }
}

// ---------------------------------------------------------------------------
// Fused rel-attention, flash style.  grid=(QLEN/128, BSZ, NHEAD), block=256.
// Each wave owns 16 query rows; loops over 16-wide key tiles.
//   AC[i,j] = (q_i + r_w_bias) . k_j
//   BD[i,j] = BDraw[i, j + QLEN-1 - i],  BDraw[i,d] = (q_i + r_r_bias) . R_d
// BDraw computed as a 16x32 WMMA band; per-row diagonal shift applied with
// ds_bpermute inside the accumulator layout.
// ---------------------------------------------------------------------------
__global__ __launch_bounds__(256)
void attn_fused(const float* __restrict__ Qh,   // [4096, 512]  row = i*BSZ+b
                const float* __restrict__ KV,   // [8192, 1024] row = j*BSZ+b; K cols 0..511, V 512..1023
                const float* __restrict__ Rh,   // [2048, 512]
                const float* __restrict__ rwb,  // [8,64]
                const float* __restrict__ rrb,  // [8,64]
                float* __restrict__ AV)         // [4096, 512]
{
    __shared__ float Plds[8][256];              // per-wave 16x16 P staging

    const int tid  = threadIdx.x;
    const int wave = tid >> 5;
    const int lane = tid & 31;
    const int ln16 = lane & 15;
    const int hi   = lane >> 4;
    const int kb   = hi << 1;
    const int bIdx = blockIdx.y;
    const int h    = blockIdx.z;
    const int i0   = blockIdx.x * 128 + wave * 16;

    // Preload A-fragments for (Q + r_w_bias) and (Q + r_r_bias), K = 64.
    v2f qw[16], qr[16];
    {
        const float2* qrow2 = (const float2*)(Qh + (size_t)((i0 + ln16) * BSZ + bIdx) * DMODEL + h * DHEAD);
        const float2* bw2   = (const float2*)(rwb + h * DHEAD);
        const float2* br2   = (const float2*)(rrb + h * DHEAD);
        #pragma unroll
        for (int f = 0; f < 16; ++f) {
            float2 q2 = qrow2[2 * f + hi];
            float2 w2 = bw2[2 * f + hi];
            float2 r2 = br2[2 * f + hi];
            qw[f].x = q2.x + w2.x;  qw[f].y = q2.y + w2.y;
            qr[f].x = q2.x + r2.x;  qr[f].y = q2.y + r2.y;
        }
    }

    v8f o0 = {0.f,0.f,0.f,0.f,0.f,0.f,0.f,0.f};
    v8f o1 = o0, o2 = o0, o3 = o0;
    float mrow[8], lrow[8];
    #pragma unroll
    for (int v = 0; v < 8; ++v) { mrow[v] = NEGBIG; lrow[v] = 0.f; }

    float* pl = &Plds[wave][0];
    const int jend = i0 + 15 + MLEN;            // last valid key index

    for (int j0 = 0; j0 <= jend; j0 += 16) {
        const float* krow = KV + (size_t)((j0 + ln16) * BSZ + bIdx) * 1024 + h * DHEAD;
        // L2 is 192MB: whole K/V panel is L2-resident; prefetch next tiles
        __builtin_prefetch(krow + 16 * BSZ * 1024, 0, 3);
        __builtin_prefetch(krow + 16 * BSZ * 1024 + 512, 0, 3);

        // ---- AC = (Q+rwb) @ K^T : 16 wmma ----
        v8f ac = {0.f,0.f,0.f,0.f,0.f,0.f,0.f,0.f};
        {
            const float2* krow2 = (const float2*)krow;
            #pragma unroll
            for (int f = 0; f < 16; ++f) {
                float2 k2 = krow2[2 * f + hi];
                v2f bf; bf.x = k2.x; bf.y = k2.y;
                ac = wmma_f32(qw[f], bf, ac);
            }
        }

        // ---- BDraw band, cols d = dbase .. dbase+31 : 32 wmma ----
        const int dbase = j0 - i0 + (QLEN - 16);     // >= 0 always
        v8f bd0 = {0.f,0.f,0.f,0.f,0.f,0.f,0.f,0.f};
        v8f bd1 = bd0;
        {
            const int d0 = dbase + ln16;             // < KLEN always
            const int d1 = dbase + 16 + ln16;        // may exceed KLEN (masked region)
            const bool ok1 = (d1 < KLEN);
            const float2* r0 = (const float2*)(Rh + (size_t)d0 * DMODEL + h * DHEAD);
            const float2* r1 = (const float2*)(Rh + (size_t)(ok1 ? d1 : 0) * DMODEL + h * DHEAD);
            #pragma unroll
            for (int f = 0; f < 16; ++f) {
                float2 x0 = r0[2 * f + hi];
                float2 x1; x1.x = 0.f; x1.y = 0.f;
                if (ok1) x1 = r1[2 * f + hi];
                v2f b0; b0.x = x0.x; b0.y = x0.y;
                v2f b1; b1.x = x1.x; b1.y = x1.y;
                bd0 = wmma_f32(qr[f], b0, bd0);
                bd1 = wmma_f32(qr[f], b1, bd1);
            }
        }

        // ---- assemble masked scores; rel-shift via bpermute ----
        float p[8], alpha[8];
        #pragma unroll
        for (int v = 0; v < 8; ++v) {
            int mi = v + (hi << 3);
            int cc = ln16 + 15 - mi;                 // 0..30: shifted BDraw column
            int srcLane = (cc & 15) | (lane & 16);   // same half, same VGPR row
            float g0 = lane_gather(bd0[v], srcLane);
            float g1 = lane_gather(bd1[v], srcLane);
            float bd = (cc < 16) ? g0 : g1;
            float s = (ac[v] + bd) * SCALE;
            if (j0 + ln16 > i0 + mi + MLEN) s = NEGBIG;   // causal mask
            p[v] = s;
        }

        // ---- online softmax (per-row over each 16-lane half) ----
        #pragma unroll
        for (int v = 0; v < 8; ++v) {
            float rm = p[v];
            rm = fmaxf(rm, __shfl_xor(rm, 1, 32));
            rm = fmaxf(rm, __shfl_xor(rm, 2, 32));
            rm = fmaxf(rm, __shfl_xor(rm, 4, 32));
            rm = fmaxf(rm, __shfl_xor(rm, 8, 32));
            float mnew = fmaxf(mrow[v], rm);
            float a  = __expf(mrow[v] - mnew);
            float pe = __expf(p[v] - mnew);
            float rs = pe;
            rs += __shfl_xor(rs, 1, 32);
            rs += __shfl_xor(rs, 2, 32);
            rs += __shfl_xor(rs, 4, 32);
            rs += __shfl_xor(rs, 8, 32);
            lrow[v] = lrow[v] * a + rs;
            mrow[v] = mnew;
            alpha[v] = a;
            p[v] = pe;
        }

        #pragma unroll
        for (int v = 0; v < 8; ++v) {
            o0[v] *= alpha[v]; o1[v] *= alpha[v];
            o2[v] *= alpha[v]; o3[v] *= alpha[v];
        }

        // ---- P through LDS (accumulator layout -> A-fragment layout) ----
        #pragma unroll
        for (int v = 0; v < 8; ++v)
            pl[(v + (hi << 3)) * 16 + ln16] = p[v];
        // same-wave LDS RAW: DS in-order; compiler inserts s_wait_dscnt

        // ---- O += P @ V : 16 wmma ----
        #pragma unroll
        for (int k0 = 0; k0 < 16; k0 += 4) {
            float2 a2 = *(const float2*)(pl + ln16 * 16 + k0 + kb);
            v2f af; af.x = a2.x; af.y = a2.y;
            const float* v0p = KV + (size_t)((j0 + k0 + kb)     * BSZ + bIdx) * 1024 + 512 + h * DHEAD;
            const float* v1p = KV + (size_t)((j0 + k0 + kb + 1) * BSZ + bIdx) * 1024 + 512 + h * DHEAD;
            v2f bf;
            bf.x = v0p[ln16];      bf.y = v1p[ln16];      o0 = wmma_f32(af, bf, o0);
            bf.x = v0p[16 + ln16]; bf.y = v1p[16 + ln16]; o1 = wmma_f32(af, bf, o1);
            bf.x = v0p[32 + ln16]; bf.y = v1p[32 + ln16]; o2 = wmma_f32(af, bf, o2);
            bf.x = v0p[48 + ln16]; bf.y = v1p[48 + ln16]; o3 = wmma_f32(af, bf, o3);
        }
    }

    // ---- epilogue: normalize and store ----
    #pragma unroll
    for (int v = 0; v < 8; ++v) {
        int mi = v + (hi << 3);
        float invl = 1.0f / lrow[v];
        float* dst = AV + (size_t)((i0 + mi) * BSZ + bIdx) * DMODEL + h * DHEAD + ln16;
        dst[0]  = o0[v] * invl;
        dst[16] = o1[v] * invl;
        dst[32] = o2[v] * invl;
        dst[48] = o3[v] * invl;
    }
}

// ---------------------------------------------------------------------------
// Residual + LayerNorm, b128 vectorized. One wave per 512-wide row.
// grid = 4096/8, block 256.
// ---------------------------------------------------------------------------
__global__ __launch_bounds__(256)
void ln_res(const float* __restrict__ w, const float* __restrict__ ao,
            const float* __restrict__ g, const float* __restrict__ bta,
            float* __restrict__ out)
{
    const int row  = blockIdx.x * 8 + (threadIdx.x >> 5);
    const int lane = threadIdx.x & 31;
    const float4* wr4 = (const float4*)(w  + (size_t)row * DMODEL);
    const float4* ar4 = (const float4*)(ao + (size_t)row * DMODEL);
    const float4* g4  = (const float4*)g;
    const float4* b4  = (const float4*)bta;

    float4 x[4];
    float s = 0.f;
    #pragma unroll
    for (int t = 0; t < 4; ++t) {
        int c = t * 32 + lane;                 // float4 index
        float4 a = wr4[c], b = ar4[c];
        x[t].x = a.x + b.x; x[t].y = a.y + b.y;
        x[t].z = a.z + b.z; x[t].w = a.w + b.w;
        s += x[t].x + x[t].y + x[t].z + x[t].w;
    }
    s += __shfl_xor(s, 1, 32);  s += __shfl_xor(s, 2, 32);
    s += __shfl_xor(s, 4, 32);  s += __shfl_xor(s, 8, 32);
    s += __shfl_xor(s, 16, 32);
    float mu = s * (1.0f / DMODEL);

    float vs = 0.f;
    #pragma unroll
    for (int t = 0; t < 4; ++t) {
        float dx = x[t].x - mu, dy = x[t].y - mu, dz = x[t].z - mu, dw = x[t].w - mu;
        vs += dx * dx + dy * dy + dz * dz + dw * dw;
    }
    vs += __shfl_xor(vs, 1, 32); vs += __shfl_xor(vs, 2, 32);
    vs += __shfl_xor(vs, 4, 32); vs += __shfl_xor(vs, 8, 32);
    vs += __shfl_xor(vs, 16, 32);
    float inv = rsqrtf(vs * (1.0f / DMODEL) + 1e-5f);

    float4* o4 = (float4*)(out + (size_t)row * DMODEL);
    #pragma unroll
    for (int t = 0; t < 4; ++t) {
        int c = t * 32 + lane;
        float4 gg = g4[c], bb = b4[c], r;
        r.x = (x[t].x - mu) * inv * gg.x + bb.x;
        r.y = (x[t].y - mu) * inv * gg.y + bb.y;
        r.z = (x[t].z - mu) * inv * gg.z + bb.z;
        r.w = (x[t].w - mu) * inv * gg.w + bb.w;
        o4[c] = r;
    }
}

// ---------------------------------------------------------------------------
extern "C" void kernel_launch(void* const* d_in, const int* in_sizes, int n_in,
                              void* d_out, int out_size, void* d_ws, size_t ws_size,
                              hipStream_t stream) {
    (void)in_sizes; (void)n_in; (void)out_size; (void)ws_size;
    const float* w    = (const float*)d_in[0];   // [1024,4,512]
    const float* r    = (const float*)d_in[1];   // [2048,1,512]
    const float* rwb  = (const float*)d_in[2];   // [8,64]
    const float* rrb  = (const float*)d_in[3];   // [8,64]
    const float* mems = (const float*)d_in[4];   // [1024,4,512]
    const float* Wq   = (const float*)d_in[5];   // [512,512]
    const float* Wkv  = (const float*)d_in[6];   // [512,1024]
    const float* Wr   = (const float*)d_in[7];   // [512,512]
    const float* Wo   = (const float*)d_in[8];   // [512,512]
    const float* lng  = (const float*)d_in[9];   // [512]
    const float* lnb  = (const float*)d_in[10];  // [512]
    // d_in[11] = attn_mask: recomputed analytically in-kernel

    float* out = (float*)d_out;
    float* wsf = (float*)d_ws;
    float* q  = wsf;                       // 4096*512
    float* kv = q  + 4096 * 512;           // 8192*1024 (K|V)
    float* rh = kv + 8192 * 1024;          // 2048*512
    float* av = rh + 2048 * 512;           // 4096*512
    float* ao = av + 4096 * 512;           // 4096*512

    const int    BIG    = 1 << 30;
    const size_t LDS_B  = 512 * 64 * sizeof(float);   // 128 KB B panel

    // Projections
    gemm512<<<dim3(4096 / 128,  512 / 64), 256, LDS_B, stream>>>(w,    w, BIG,  Wq,  q,  512);
    gemm512<<<dim3(8192 / 128, 1024 / 64), 256, LDS_B, stream>>>(mems, w, 4096, Wkv, kv, 1024);
    gemm512<<<dim3(2048 / 128,  512 / 64), 256, LDS_B, stream>>>(r,    r, BIG,  Wr,  rh, 512);
    // Fused relative attention
    attn_fused<<<dim3(QLEN / 128, BSZ, NHEAD), 256, 0, stream>>>(q, kv, rh, rwb, rrb, av);
    // Output projection + residual LayerNorm
    gemm512<<<dim3(4096 / 128, 512 / 64), 256, LDS_B, stream>>>(av, av, BIG, Wo, ao, 512);
    ln_res<<<dim3(4096 / 8), 256, 0, stream>>>(w, ao, lng, lnb, out);
}